// QueryAndGroup_68436008895055
// MI455X (gfx1250) — compile-verified
//
#include <hip/hip_runtime.h>

typedef __attribute__((ext_vector_type(2))) float v2f;
typedef __attribute__((ext_vector_type(8))) float v8f;

#define B_   4
#define N_   8192
#define S_   2048
#define C_   64
#define NS_  32
#define R2_  0.04f
#define WPB  4   // waves per block (wave32)

#define XYZ_LDS_BYTES (N_ * 12)   // 96 KB packed (x,y,z) per batch, async-staged

__global__ __launch_bounds__(WPB * 32)
void QueryAndGroup_wmma_kernel(const float* __restrict__ xyz,
                               const float* __restrict__ new_xyz,
                               const float* __restrict__ feat,
                               float* __restrict__ out)
{
    extern __shared__ char smem[];                 // dynamic: 96 KB xyz stage
    float* sxyz = (float*)smem;
    __shared__ int idxBuf[WPB][16][NS_];           // static: 8 KB

    const int lane = threadIdx.x & 31;
    const int w    = threadIdx.x >> 5;
    const int tile = blockIdx.x * WPB + w;         // one 16-query tile per wave
    const int b    = tile / (S_ / 16);             // all waves in block share b
    const int s0   = (tile % (S_ / 16)) * 16;
    const bool hi  = lane >= 16;
    const int  m   = lane & 15;

    // ================= async-stage batch xyz (96 KB) into LDS =================
    {
        const unsigned ldsBase = (unsigned)(size_t)sxyz;   // low 32b of flat = LDS offset
        const float* gbase = xyz + (size_t)b * N_ * 3;
        for (int i = threadIdx.x; i < (XYZ_LDS_BYTES / 16); i += WPB * 32) {
            unsigned lofs = ldsBase + (unsigned)i * 16u;
            const float* g = gbase + (size_t)i * 4;
            asm volatile("global_load_async_to_lds_b128 %0, %1, off"
                         :: "v"(lofs), "v"(g) : "memory");
        }
        asm volatile("s_wait_asynccnt 0x0" ::: "memory");
    }
    __syncthreads();

    // ====== A matrix: row m = (-2qx,-2qy,-2qz, 1); lanes0-15 K0/K1, lanes16-31 K2/K3
    const float* q = new_xyz + ((size_t)b * S_ + s0 + m) * 3;
    const float qx = q[0], qy = q[1], qz = q[2];
    const float q2v = qx * qx + qy * qy + qz * qz;
    v2f a;
    a.x = hi ? (-2.0f * qz) : (-2.0f * qx);
    a.y = hi ? 1.0f         : (-2.0f * qy);

    // C accumulator: C[i] = q^2 of row (i + 8*hi) via lane permute
    v8f cacc;
#pragma unroll
    for (int i = 0; i < 8; ++i) {
        int src = i + (hi ? 8 : 0);
        cacc[i] = __int_as_float(
            __builtin_amdgcn_ds_bpermute(src << 2, __float_as_int(q2v)));
    }

    int cnt = hi ? NS_ : 0;   // lanes 0-15 own rows 0-15; others start "full"

    // ================= sweep: 32 points (2 WMMAs) per iteration ===============
    for (int base = 0; base < N_; base += 32) {
        const int o0 = (base + m) * 3;
        const float ax0 = sxyz[o0], ay0 = sxyz[o0 + 1], az0 = sxyz[o0 + 2];
        const int o1 = (base + 16 + m) * 3;
        const float ax1 = sxyz[o1], ay1 = sxyz[o1 + 1], az1 = sxyz[o1 + 2];
        const float pp0 = ax0 * ax0 + ay0 * ay0 + az0 * az0;
        const float pp1 = ax1 * ax1 + ay1 * ay1 + az1 * az1;

        v2f bv0, bv1;
        bv0.x = hi ? az0 : ax0;  bv0.y = hi ? pp0 : ay0;
        bv1.x = hi ? az1 : ax1;  bv1.y = hi ? pp1 : ay1;

        v8f d0 = __builtin_amdgcn_wmma_f32_16x16x4_f32(
            false, a, false, bv0, (short)0, cacc, false, false);
        v8f d1 = __builtin_amdgcn_wmma_f32_16x16x4_f32(
            false, a, false, bv1, (short)0, cacc, false, false);

        unsigned mA[8], mB[8];
#pragma unroll
        for (int i = 0; i < 8; ++i) {
            mA[i] = __builtin_amdgcn_ballot_w32(d0[i] < R2_);
            mB[i] = __builtin_amdgcn_ballot_w32(d1[i] < R2_);
        }

        // lane r selects mask of its row: vgpr index = r&7, half = r>>3
        unsigned a0 = (lane & 1) ? mA[1] : mA[0];
        unsigned a1 = (lane & 1) ? mA[3] : mA[2];
        unsigned a2 = (lane & 1) ? mA[5] : mA[4];
        unsigned a3 = (lane & 1) ? mA[7] : mA[6];
        unsigned b0 = (lane & 1) ? mB[1] : mB[0];
        unsigned b1 = (lane & 1) ? mB[3] : mB[2];
        unsigned b2 = (lane & 1) ? mB[5] : mB[4];
        unsigned b3 = (lane & 1) ? mB[7] : mB[6];
        unsigned au0 = (lane & 2) ? a1 : a0;
        unsigned au1 = (lane & 2) ? a3 : a2;
        unsigned bu0 = (lane & 2) ? b1 : b0;
        unsigned bu1 = (lane & 2) ? b3 : b2;
        unsigned seA = (lane & 4) ? au1 : au0;
        unsigned seB = (lane & 4) ? bu1 : bu0;
        unsigned m16A = ((lane & 8) ? (seA >> 16) : seA) & 0xFFFFu;
        unsigned m16B = ((lane & 8) ? (seB >> 16) : seB) & 0xFFFFu;
        unsigned mask32 = m16A | (m16B << 16);   // bit k <-> point base+k, in order

        while (mask32 && cnt < NS_) {
            int bit = __builtin_ctz(mask32);
            mask32 &= mask32 - 1;
            idxBuf[w][m][cnt] = base + bit;
            ++cnt;
        }

        if (__builtin_amdgcn_ballot_w32(cnt >= NS_) == 0xFFFFFFFFu) break;
    }

    // pad with first hit (or 0 if no hit) — reference semantics
    if (!hi) {
        int first = (cnt > 0) ? idxBuf[w][m][0] : 0;
        for (int k = cnt; k < NS_; ++k) idxBuf[w][m][k] = first;
    }
    // same-wave LDS ordering: no barrier needed before our own reads

    // ================= gather + write: 67 coalesced 128B rows per query =======
    const size_t chs = (size_t)S_ * NS_;
#pragma unroll 1
    for (int r = 0; r < 16; ++r) {
        const int s  = s0 + r;
        const int id = idxBuf[w][r][lane];
        const float px = sxyz[id * 3], py = sxyz[id * 3 + 1], pz = sxyz[id * 3 + 2];
        const float* qq = new_xyz + ((size_t)b * S_ + s) * 3;
        const size_t ob = (((size_t)b * 67) * S_ + s) * NS_ + lane;
        out[ob + 0 * chs] = px - qq[0];
        out[ob + 1 * chs] = py - qq[1];
        out[ob + 2 * chs] = pz - qq[2];
        const float* fb = feat + ((size_t)b * C_) * N_ + id;
#pragma unroll 4
        for (int c = 0; c < C_; ++c)
            out[ob + (size_t)(3 + c) * chs] = fb[(size_t)c * N_];
    }
}

extern "C" void kernel_launch(void* const* d_in, const int* in_sizes, int n_in,
                              void* d_out, int out_size, void* d_ws, size_t ws_size,
                              hipStream_t stream)
{
    (void)in_sizes; (void)n_in; (void)out_size; (void)d_ws; (void)ws_size;
    const float* xyz  = (const float*)d_in[0];
    const float* nxyz = (const float*)d_in[1];
    const float* feat = (const float*)d_in[2];
    float* out = (float*)d_out;

    const int tiles  = B_ * (S_ / 16);   // 512 query tiles, one per wave
    const int blocks = tiles / WPB;      // 128 blocks of 4 waves
    QueryAndGroup_wmma_kernel<<<blocks, WPB * 32, XYZ_LDS_BYTES, stream>>>(
        xyz, nxyz, feat, out);
}